// NodeAttention_42666205119202
// MI455X (gfx1250) — compile-verified
//
#include <hip/hip_runtime.h>
#include <stdint.h>

// Problem dims (fixed by the reference): x[B,S,V,D]
#define B_ 2
#define S_ 128
#define V_ 1024
#define D_ 512

typedef __attribute__((ext_vector_type(2))) float v2f;
typedef __attribute__((ext_vector_type(8))) float v8f;
typedef __attribute__((ext_vector_type(4))) unsigned int u32x4;
typedef __attribute__((ext_vector_type(8))) int i32x8;
typedef __attribute__((ext_vector_type(4))) int i32x4;

// ---------------------------------------------------------------------------
// Kernel 1: collapse the two linear layers (no nonlinearity between them):
//   veff[d] = sum_k W2[0,k] * W1[k,d]       (512 floats)
//   c       = sum_k W2[0,k] * b1[k] + b2[0] (1 float, stored at ws[512])
// ---------------------------------------------------------------------------
__global__ void prep_veff(const float* __restrict__ W1, const float* __restrict__ b1,
                          const float* __restrict__ W2, const float* __restrict__ b2,
                          float* __restrict__ ws) {
  const int d = threadIdx.x;  // 512 threads
  float acc = 0.f;
#pragma unroll
  for (int k = 0; k < 32; ++k) acc += W2[k] * W1[k * D_ + d];
  ws[d] = acc;
  if (d == 0) {
    float c = 0.f;
#pragma unroll
    for (int k = 0; k < 32; ++k) c += W2[k] * b1[k];
    ws[D_] = c + b2[0];
  }
}

// Issue one TDM 2-D tile load: `rows` lines of 512 f32, line stride V*D elems.
__device__ __forceinline__ void tdm_load_rows(const float* gsrc, unsigned int ldsOff,
                                              int rows) {
  const unsigned long long ga = (unsigned long long)(uintptr_t)gsrc;  // byte addr
  u32x4 g0;
  g0[0] = 1u;                                        // count=1, user D#, no gather
  g0[1] = ldsOff;                                    // lds_addr (bytes)
  g0[2] = (unsigned int)(ga & 0xffffffffull);        // global_addr lo
  g0[3] = (unsigned int)(ga >> 32) | (2u << 30);     // global_addr hi | type=2
  i32x8 g1;
  g1[0] = 0x00020000;                                // data_size=2 (4B), no mask/pad
  g1[1] = (int)(((unsigned)D_) << 16);               // tensor_dim0 = 512
  g1[2] = (int)(((unsigned)rows) << 16);             // tensor_dim1 = rows
  g1[3] = (int)(((unsigned)D_) << 16);               // tile_dim0 = 512
  g1[4] = rows;                                      // tile_dim1 = rows, tile_dim2 = 0
  g1[5] = V_ * D_;                                   // tensor_dim0_stride = 524288 elems
  g1[6] = 0;
  g1[7] = 0;
  i32x4 z4 = {0, 0, 0, 0};                           // groups 2/3 unused (2-D tensor)
  i32x8 z8 = {0, 0, 0, 0, 0, 0, 0, 0};               // extra group (clang-23 6-arg form)
  __builtin_amdgcn_tensor_load_to_lds(g0, g1, z4, z4, z8, 0);
}

// ---------------------------------------------------------------------------
// Kernel 2: one workgroup per (b,v).
//   Two TDM loads of x[b,:,v,:] halves (64x512 f32 = 128KB each) into LDS,
//   staged s_wait_tensorcnt to overlap DMA of half B with scores on half A,
//   then softmax over s and weighted sum via V_WMMA_F32_16X16X4_F32.
// ---------------------------------------------------------------------------
__launch_bounds__(512, 1)
__global__ void node_attn(const float* __restrict__ x,
                          const float* __restrict__ veff_c,
                          float* __restrict__ out) {
  __shared__ float xT[S_ * D_];  // 256 KB tile: xT[s*512 + d]
  __shared__ float sV[D_];       // veff
  __shared__ float sW[S_];       // scores, then softmax weights

  const int tid  = threadIdx.x;
  const int lane = tid & 31;
  const int wv   = tid >> 5;      // wave id 0..15
  const int bv   = blockIdx.x;    // 0..2047
  const int b    = bv >> 10;
  const int v    = bv & (V_ - 1);

  // x[b, s, v, d] flat offset = ((b*S + s)*V + v)*D + d ; row (s) stride = V*D
  const float* xbase = x + ((size_t)b * S_ * V_ + (size_t)v) * D_;

  // ---- Wave 0 kicks off two TDM loads (rows 0-63, rows 64-127) ----
  if (wv == 0) {
    const unsigned int ldsOff = (unsigned int)(uintptr_t)&xT[0];
    tdm_load_rows(xbase, ldsOff, 64);
    tdm_load_rows(xbase + (size_t)64 * V_ * D_, ldsOff + 64u * D_ * 4u, 64);
  }

  // veff -> LDS while the DMA is in flight (blockDim == D_ == 512)
  sV[tid] = veff_c[tid];
  const float cval = veff_c[D_];

  // ---- First half arrived (TENSORcnt is in-order): score rows 0..63 ----
  if (wv == 0) __builtin_amdgcn_s_wait_tensorcnt(1);
  __syncthreads();

#pragma unroll
  for (int r = wv * 4; r < wv * 4 + 4; ++r) {   // 16 waves x 4 rows = 64
    const float* row = &xT[r * D_];
    float acc = 0.f;
#pragma unroll 4
    for (int i = lane; i < D_; i += 32) acc += row[i] * sV[i];  // conflict-free
#pragma unroll
    for (int m = 16; m >= 1; m >>= 1) acc += __shfl_xor(acc, m, 32);
    if (lane == 0) sW[r] = acc + cval;
  }

  // ---- Second half: wait remaining DMA, score rows 64..127 ----
  if (wv == 0) __builtin_amdgcn_s_wait_tensorcnt(0);
  __syncthreads();

#pragma unroll
  for (int r = 64 + wv * 4; r < 64 + wv * 4 + 4; ++r) {
    const float* row = &xT[r * D_];
    float acc = 0.f;
#pragma unroll 4
    for (int i = lane; i < D_; i += 32) acc += row[i] * sV[i];
#pragma unroll
    for (int m = 16; m >= 1; m >>= 1) acc += __shfl_xor(acc, m, 32);
    if (lane == 0) sW[r] = acc + cval;
  }
  __syncthreads();

  // ---- Softmax over s (single wave; 4 scores per lane) ----
  if (wv == 0) {
    float s0 = sW[lane], s1 = sW[lane + 32], s2 = sW[lane + 64], s3 = sW[lane + 96];
    float mx = fmaxf(fmaxf(s0, s1), fmaxf(s2, s3));
#pragma unroll
    for (int m = 16; m >= 1; m >>= 1) mx = fmaxf(mx, __shfl_xor(mx, m, 32));
    float e0 = __expf(s0 - mx), e1 = __expf(s1 - mx);
    float e2 = __expf(s2 - mx), e3 = __expf(s3 - mx);
    float sum = e0 + e1 + e2 + e3;
#pragma unroll
    for (int m = 16; m >= 1; m >>= 1) sum += __shfl_xor(sum, m, 32);
    const float inv = 1.f / sum;
    sW[lane] = e0 * inv; sW[lane + 32] = e1 * inv;
    sW[lane + 64] = e2 * inv; sW[lane + 96] = e3 * inv;
  }
  __syncthreads();

  // ---- Pass 2 (WMMA f32 16x16x4): out[d0..d0+15] = sum_s w[s]*xT[s][d0..].
  // A[m][k] = w[s0+k] for all m (broadcast rows); B[k][n] = xT[s0+k][d0+n].
  // ISA layout: lanes 0-15 carry K=0,1 in VGPR0,1; lanes 16-31 carry K=2,3.
  float* obase = out + ((size_t)b * V_ + (size_t)v) * D_;
  const int n  = lane & 15;
  const int ko = (lane >> 4) << 1;  // 0 for lanes 0-15, 2 for lanes 16-31
  for (int t = wv; t < 32; t += 16) {  // 32 d-tiles, 2 per wave (uniform trip count)
    const int d0 = t * 16;
    v8f acc = {};
    for (int s0 = 0; s0 < S_; s0 += 4) {
      const int sA = s0 + ko;
      v2f a;  a[0]  = sW[sA];                 a[1]  = sW[sA + 1];
      v2f bb; bb[0] = xT[sA * D_ + d0 + n];   bb[1] = xT[(sA + 1) * D_ + d0 + n];
      acc = __builtin_amdgcn_wmma_f32_16x16x4_f32(
          /*neg_a=*/false, a, /*neg_b=*/false, bb,
          /*c_mod=*/(short)0, acc, /*reuse_a=*/false, /*reuse_b=*/false);
    }
    // D row M=0 (lanes 0-15, VGPR0) is the result; rows 1-15 are duplicates.
    if (lane < 16) obase[d0 + lane] = acc[0];
  }
}

// ---------------------------------------------------------------------------
extern "C" void kernel_launch(void* const* d_in, const int* in_sizes, int n_in,
                              void* d_out, int out_size, void* d_ws, size_t ws_size,
                              hipStream_t stream) {
  const float* x  = (const float*)d_in[0];
  const float* W1 = (const float*)d_in[1];
  const float* b1 = (const float*)d_in[2];
  const float* W2 = (const float*)d_in[3];
  const float* b2 = (const float*)d_in[4];
  float* out = (float*)d_out;
  float* ws  = (float*)d_ws;  // needs 513 floats

  prep_veff<<<1, 512, 0, stream>>>(W1, b1, W2, b2, ws);
  node_attn<<<B_ * V_, 512, 0, stream>>>(x, ws, out);
}